// SelfCorrelationBlock_32719060861401
// MI455X (gfx1250) — compile-verified
//
#include <hip/hip_runtime.h>
#include <math.h>

// CDNA5 / gfx1250, wave32.
typedef __attribute__((ext_vector_type(2))) float v2f;
typedef __attribute__((ext_vector_type(8))) float v8f;

#define Hdim 96
#define Wdim 96
#define HW   (Hdim * Wdim)
#define Cdim 128
#define Rdim 32
#define TILE 16
#define PADW 3
#define HALO 24              // 22 rows/cols needed (16 + 2*3), padded to 24
#define NP   (HALO * HALO)   // 576 halo pixels
#define FSTR 584             // feat LDS row stride (576+8, bank-conflict pad)
#define CSTR 264             // corr LDS row stride (256+8)
#define WRSTR 130            // W_reduce LDS row stride (128+2, even for b64 reads)
#define WPSTR 34             // W_proj  LDS row stride (32+2)
#define EPSV 1e-8f

#define WBUF_OFF 0
#define WBUF_FLOATS 4352                  // max(32*130, 128*34)
#define FEAT_OFF WBUF_FLOATS
#define CORR_OFF (FEAT_OFF + Rdim * FSTR)
#define SMEM_FLOATS (CORR_OFF + Rdim * CSTR)   // 31488 floats = 125952 B < 320 KB/WGP

__device__ __forceinline__ v8f wmma_f32_k4(v2f a, v2f b, v8f c) {
  // D = A(16x4 f32) * B(4x16 f32) + C(16x16 f32)
  return __builtin_amdgcn_wmma_f32_16x16x4_f32(
      /*neg_a=*/false, a, /*neg_b=*/false, b,
      /*c_mod=*/(short)0, c, /*reuse_a=*/false, /*reuse_b=*/false);
}

__global__ __launch_bounds__(256) void selfcorr_fused_kernel(
    const float* __restrict__ x, const float* __restrict__ Wr,
    const float* __restrict__ Wp, float* __restrict__ out) {
  extern __shared__ float smem[];
  float* wbuf = smem + WBUF_OFF;   // W_reduce, later reused for W_proj
  float* feat = smem + FEAT_OFF;   // [32][FSTR] halo feature tile
  float* corr = smem + CORR_OFF;   // [32][CSTR] attended features

  const int tid  = threadIdx.x;
  const int lane = tid & 31;
  const int wv   = tid >> 5;       // 8 waves per workgroup
  const int ln   = lane & 15;      // column-within-tile lane group
  const int hi   = lane >> 4;      // 0: K low half, 1: K high half

  const int bz  = blockIdx.y;
  const int ty0 = (blockIdx.x / (Wdim / TILE)) * TILE;
  const int tx0 = (blockIdx.x % (Wdim / TILE)) * TILE;
  const long xb = (long)bz * Cdim * HW;

  // ---- stage W_reduce (32x128) into LDS, padded stride ----
  for (int i = tid; i < Rdim * Cdim; i += 256)
    wbuf[(i >> 7) * WRSTR + (i & 127)] = Wr[i];
  __syncthreads();

  // ---- Phase 2: feat(32 x 576) = W_reduce(32x128) @ x_halo(128x576) via f32 WMMA ----
  // 2 M-tiles x 36 N-tiles of 16x16, round-robined over 8 waves.
  for (int t = wv; t < 2 * (NP / 16); t += 8) {
    const int m0 = (t & 1) * 16;
    const int n0 = (t >> 1) * 16;

    // Boundary handling is per-COLUMN (constant over K): clamp address, 0/1 mask.
    // EXEC stays all-1s -> unconditional loads, no saveexec branches.
    const int p  = n0 + ln;
    const int hy = p / HALO, hx = p - hy * HALO;
    const int gy = ty0 - PADW + hy;
    const int gx = tx0 - PADW + hx;
    const bool inb = ((unsigned)gy < (unsigned)Hdim) && ((unsigned)gx < (unsigned)Wdim);
    const float msk = inb ? 1.0f : 0.0f;
    // One 64-bit base address; channel offsets become compile-time immediates.
    const float* __restrict__ xq =
        x + xb + (long)(inb ? gy : 0) * Wdim + (inb ? gx : 0) + 2 * hi * HW;

    // Issue all 64 B-loads first (software pipelining), then 32 back-to-back WMMAs.
    float bx[Cdim / 4], by[Cdim / 4];
#pragma unroll
    for (int ks = 0; ks < Cdim / 4; ++ks) {
      bx[ks] = xq[ks * 4 * HW];
      by[ks] = xq[ks * 4 * HW + HW];
    }
    // All 32 A-frags share one LDS address; 16B-stride DS immediate offsets.
    const float* __restrict__ aq = &wbuf[(m0 + ln) * WRSTR + 2 * hi];
    v8f acc = {};
#pragma unroll
    for (int ks = 0; ks < Cdim / 4; ++ks) {
      v2f a = *(const v2f*)&aq[ks * 4];
      v2f b;
      b.x = bx[ks] * msk;
      b.y = by[ks] * msk;
      acc = wmma_f32_k4(a, b, acc);
    }
    // D layout: VGPR v -> row (v + 8*hi), col = ln
#pragma unroll
    for (int v = 0; v < 8; ++v)
      feat[(m0 + v + 8 * hi) * FSTR + n0 + ln] = acc[v];
  }
  __syncthreads();

  // ---- Phase 3: cosine-sim + online softmax + weighted sum (one thread per pixel) ----
  {
    const int iy = tid >> 4, ix = tid & 15;
    const int pc = (iy + PADW) * HALO + (ix + PADW);

    float fc[Rdim];
    float nc2 = 0.f;
#pragma unroll
    for (int r = 0; r < Rdim; ++r) {
      fc[r] = feat[r * FSTR + pc];
      nc2 += fc[r] * fc[r];
    }
    const float ncn = sqrtf(nc2);

    float mrun = -3.0e38f, ssum = 0.f;
    float cr[Rdim];
#pragma unroll
    for (int r = 0; r < Rdim; ++r) cr[r] = 0.f;

    for (int k = 0; k < 49; ++k) {          // window offsets, row-major (dy,dx)
      const int dy = k / 7, dx = k % 7;
      const int pp = (iy + dy) * HALO + (ix + dx);
      float fp[Rdim];
      float dot = 0.f, nn = 0.f;
#pragma unroll
      for (int r = 0; r < Rdim; ++r) {
        const float v = feat[r * FSTR + pp];
        fp[r] = v;
        dot += fc[r] * v;
        nn += v * v;
      }
      const float sim = dot / fmaxf(ncn * sqrtf(nn), EPSV);
      if (sim > mrun) {                     // online softmax rescale
        const float sc = expf(mrun - sim);
        ssum *= sc;
#pragma unroll
        for (int r = 0; r < Rdim; ++r) cr[r] *= sc;
        mrun = sim;
      }
      const float w = expf(sim - mrun);
      ssum += w;
#pragma unroll
      for (int r = 0; r < Rdim; ++r) cr[r] += w * fp[r];
    }
    const float inv = 1.0f / ssum;
#pragma unroll
    for (int r = 0; r < Rdim; ++r) corr[r * CSTR + tid] = cr[r] * inv;
  }

  // ---- stage W_proj (128x32) into wbuf (W_reduce dead after phase 2) ----
  for (int i = tid; i < Cdim * Rdim; i += 256)
    wbuf[(i >> 5) * WPSTR + (i & 31)] = Wp[i];
  __syncthreads();

  // ---- Phase 4: out = x + W_proj(128x32) @ corr(32x256) via f32 WMMA ----
  // 8 M-tiles x 16 N-tiles of 16x16, K=32 in 8 steps.
  for (int t = wv; t < 8 * 16; t += 8) {
    const int m0 = (t & 7) * 16;            // output channel base
    const int n0 = (t >> 3) * 16;           // pixel base
    const float* __restrict__ aq = &wbuf[(m0 + ln) * WPSTR + 2 * hi];
    v8f acc = {};
#pragma unroll
    for (int ks = 0; ks < Rdim / 4; ++ks) {
      const int kb = ks * 4 + 2 * hi;
      v2f a = *(const v2f*)&aq[ks * 4];
      v2f b;
      b.x = corr[kb * CSTR + n0 + ln];
      b.y = corr[(kb + 1) * CSTR + n0 + ln];
      acc = wmma_f32_k4(a, b, acc);
    }
    const int pixel = n0 + ln;
    const int gy = ty0 + (pixel >> 4);
    const int gx = tx0 + (pixel & 15);
    // One base address; per-VGPR channel offsets fold into immediates.
    const long gbase = xb + (long)(m0 + 8 * hi) * HW + (long)gy * Wdim + gx;
    const float* __restrict__ xr = x + gbase;
    float* __restrict__ orp = out + gbase;
#pragma unroll
    for (int v = 0; v < 8; ++v)
      orp[v * HW] = xr[v * HW] + acc[v];    // fused residual add
  }
}

extern "C" void kernel_launch(void* const* d_in, const int* in_sizes, int n_in,
                              void* d_out, int out_size, void* d_ws, size_t ws_size,
                              hipStream_t stream) {
  const float* x  = (const float*)d_in[0];
  const float* Wr = (const float*)d_in[1];  // (32, 128) row-major
  const float* Wp = (const float*)d_in[2];  // (128, 32) row-major
  float* out = (float*)d_out;

  const size_t smem_bytes = (size_t)SMEM_FLOATS * sizeof(float);
  (void)hipFuncSetAttribute((const void*)selfcorr_fused_kernel,
                            hipFuncAttributeMaxDynamicSharedMemorySize,
                            (int)smem_bytes);

  dim3 grid((Hdim / TILE) * (Wdim / TILE), 4);  // 36 tiles x 4 batch
  selfcorr_fused_kernel<<<grid, 256, smem_bytes, stream>>>(x, Wr, Wp, out);
}